// MoLoRARouter_9990093931085
// MI455X (gfx1250) — compile-verified
//
#include <hip/hip_runtime.h>
#include <hip/hip_bf16.h>

// CDNA5 (gfx1250) MoE router: fp32 WMMA GEMM (tokens x experts) + fused top-2 epilogue.
// D = A(16x4 f32, tokens x K) * B(4x16 f32, K x experts) + C(16x16 f32), wave32.

typedef __attribute__((ext_vector_type(2))) float v2f;
typedef __attribute__((ext_vector_type(8))) float v8f;

namespace {
constexpr int kH = 4096;              // hidden size (K)
constexpr int kE = 64;                // experts (N)
constexpr int kTokens = 4 * 4096;     // B*S = 16384 (M)
constexpr int kTokensPerWave = 16;
constexpr int kWavesPerBlock = 4;
constexpr int kTokensPerBlock = kTokensPerWave * kWavesPerBlock;  // 64
}

__global__ __launch_bounds__(128, 2) void moe_router_topk_kernel(
    const float* __restrict__ x,       // [kTokens, kH]
    const float* __restrict__ gate_w,  // [kE, kH]
    float* __restrict__ out) {         // [2*kTokens weights][2*kTokens indices(as float)]
  // Padded rows (65) so the per-token column scan is bank-conflict free.
  __shared__ float logits_lds[kWavesPerBlock][kTokensPerWave][kE + 1];

  const int tid      = threadIdx.x;
  const int lane     = tid & 31;
  const int wave     = tid >> 5;
  const int laneHalf = lane >> 4;   // 0: K={k,k+1}, 1: K={k+2,k+3}
  const int l16      = lane & 15;   // A: token row (M), B: expert col (N)
  const int tok_base = blockIdx.x * kTokensPerBlock + wave * kTokensPerWave;

  // A fragment source: x[tok_base + l16][k + 2*laneHalf .. +1]
  const float* a_ptr = x + (size_t)(tok_base + l16) * kH + 2 * laneHalf;
  // B fragment source (expert group g): gate_w[g*16 + l16][k + 2*laneHalf .. +1]
  const float* b_ptr = gate_w + (size_t)l16 * kH + 2 * laneHalf;

  v8f acc0 = {}, acc1 = {}, acc2 = {}, acc3 = {};

#pragma unroll 4
  for (int k = 0; k < kH; k += 4) {
    v2f a  = *(const v2f*)(a_ptr + k);
    v2f b0 = *(const v2f*)(b_ptr + 0 * 16 * kH + k);
    v2f b1 = *(const v2f*)(b_ptr + 1 * 16 * kH + k);
    v2f b2 = *(const v2f*)(b_ptr + 2 * 16 * kH + k);
    v2f b3 = *(const v2f*)(b_ptr + 3 * 16 * kH + k);
    // 8 args: (neg_a, A, neg_b, B, c_mod, C, reuse_a, reuse_b)
    acc0 = __builtin_amdgcn_wmma_f32_16x16x4_f32(false, a, false, b0, (short)0, acc0, false, false);
    acc1 = __builtin_amdgcn_wmma_f32_16x16x4_f32(false, a, false, b1, (short)0, acc1, false, false);
    acc2 = __builtin_amdgcn_wmma_f32_16x16x4_f32(false, a, false, b2, (short)0, acc2, false, false);
    acc3 = __builtin_amdgcn_wmma_f32_16x16x4_f32(false, a, false, b3, (short)0, acc3, false, false);
  }

  // C/D layout: VGPR i, lanes 0-15 -> (M=i, N=lane); lanes 16-31 -> (M=i+8, N=lane-16).
  // Scatter logits[token][expert] into LDS.
#pragma unroll
  for (int i = 0; i < 8; ++i) {
    const int m = i + 8 * laneHalf;
    logits_lds[wave][m][0 * 16 + l16] = acc0[i];
    logits_lds[wave][m][1 * 16 + l16] = acc1[i];
    logits_lds[wave][m][2 * 16 + l16] = acc2[i];
    logits_lds[wave][m][3 * 16 + l16] = acc3[i];
  }

  __syncthreads();

  // Lanes 0..15 of each wave: one token each. Top-2 of logits == top-2 of softmax;
  // renormalized weights reduce to a 2-way sigmoid (softmax denominator cancels).
  if (lane < kTokensPerWave) {
    const float* row = &logits_lds[wave][lane][0];
    float best1 = -3.402823466e38f, best2 = -3.402823466e38f;
    int i1 = 0, i2 = 0;
    for (int e = 0; e < kE; ++e) {
      const float v = row[e];
      if (v > best1) {            // strict '>': first index wins ties (jax top_k order)
        best2 = best1; i2 = i1;
        best1 = v;     i1 = e;
      } else if (v > best2) {
        best2 = v;     i2 = e;
      }
    }
    const float w1 = 1.0f / (1.0f + __expf(best2 - best1));
    const float w2 = 1.0f - w1;
    const int t = tok_base + lane;
    out[2 * t + 0] = w1;
    out[2 * t + 1] = w2;
    out[2 * kTokens + 2 * t + 0] = (float)i1;
    out[2 * kTokens + 2 * t + 1] = (float)i2;
  }
}

extern "C" void kernel_launch(void* const* d_in, const int* in_sizes, int n_in,
                              void* d_out, int out_size, void* d_ws, size_t ws_size,
                              hipStream_t stream) {
  (void)in_sizes; (void)n_in; (void)out_size; (void)d_ws; (void)ws_size;
  const float* x      = (const float*)d_in[0];
  const float* gate_w = (const float*)d_in[1];
  float* out          = (float*)d_out;

  dim3 grid(kTokens / kTokensPerBlock);  // 256 blocks
  dim3 block(kWavesPerBlock * 32);       // 128 threads = 4 waves
  moe_router_topk_kernel<<<grid, block, 0, stream>>>(x, gate_w, out);
}